// GraphConvInstanceGlobalMaxSmall_10874857193951
// MI455X (gfx1250) — compile-verified
//
#include <hip/hip_runtime.h>
#include <hip/hip_bf16.h>
#include <math.h>

// ---------------------------------------------------------------------------
// Problem constants (from reference)
// ---------------------------------------------------------------------------
#define N_NODES 50000
#define N_EDGES 600000
#define IN_CH   31
#define HID     128
#define N_GRAPHS 512
#define N_TILES (N_NODES / 16)   // 3125 exactly
#define BIG_GRID 512             // grid-stride blocks for the TDM conv

typedef float v2f __attribute__((ext_vector_type(2)));
typedef float v8f __attribute__((ext_vector_type(8)));
typedef unsigned int u32x4 __attribute__((ext_vector_type(4)));
typedef int i32x4 __attribute__((ext_vector_type(4)));
typedef int i32x8 __attribute__((ext_vector_type(8)));

// ---------------------------------------------------------------------------
// Utility kernels
// ---------------------------------------------------------------------------
__global__ void zero_f32(float* __restrict__ p, int n) {
    int i = blockIdx.x * blockDim.x + threadIdx.x;
    if (i < n) p[i] = 0.0f;
}

// Fragment-major B layout for V_WMMA_F32_16X16X4_F32:
//   Wf[((nt*panels + p)*32 + lane)*2 + j] = B(k = 4p + 2*(lane>>4) + j,
//                                            n = nt*16 + (lane&15))
// where B = [W_rel ; W_root]^T (K x 128), zero-padded K.
__global__ void build_wfrag(const float* __restrict__ Wrel,
                            const float* __restrict__ Wroot,
                            float* __restrict__ Wf, int CIN, int panels) {
    int idx = blockIdx.x * blockDim.x + threadIdx.x;
    int total = 8 * panels * 64;
    if (idx >= total) return;
    int j    = idx & 1;
    int lane = (idx >> 1) & 31;
    int p    = (idx >> 6) % panels;
    int nt   = idx / (panels * 64);
    int k    = 4 * p + 2 * (lane >> 4) + j;
    int col  = nt * 16 + (lane & 15);
    float v = 0.0f;
    if (k < CIN)          v = Wrel[col * CIN + k];
    else if (k < 2 * CIN) v = Wroot[col * CIN + (k - CIN)];
    Wf[idx] = v;
}

// Transpose W[N][K] -> T[K][N] (for the MLP head)
__global__ void transpose_w(const float* __restrict__ W, float* __restrict__ T,
                            int N, int K) {
    int idx = blockIdx.x * blockDim.x + threadIdx.x;
    if (idx >= N * K) return;
    int k = idx / N, n = idx % N;
    T[idx] = W[n * K + k];
}

// ---------------------------------------------------------------------------
// Edge scatter-add
// ---------------------------------------------------------------------------
__global__ void scatter_add31(const float* __restrict__ x,
                              const int* __restrict__ src,
                              const int* __restrict__ dst,
                              float* __restrict__ agg) {
    long long idx = (long long)blockIdx.x * blockDim.x + threadIdx.x;
    long long total = (long long)N_EDGES * IN_CH;
    if (idx >= total) return;
    int e = (int)(idx / IN_CH);
    int c = (int)(idx % IN_CH);
    int s = src[e], d = dst[e];
    atomicAdd(&agg[(long long)d * IN_CH + c], x[(long long)s * IN_CH + c]);
}

// 128 channels: one thread per (edge, 4-channel group): float4 gather + 4 atomics
__global__ void scatter_add128(const float* __restrict__ x,
                               const int* __restrict__ src,
                               const int* __restrict__ dst,
                               float* __restrict__ agg) {
    long long idx = (long long)blockIdx.x * blockDim.x + threadIdx.x;
    long long total = (long long)N_EDGES * 32;
    if (idx >= total) return;
    int e  = (int)(idx >> 5);
    int c4 = (int)(idx & 31) << 2;
    int s = src[e], d = dst[e];
    const float4 v = *reinterpret_cast<const float4*>(&x[(long long)s * HID + c4]);
    float* base = &agg[(long long)d * HID + c4];
    atomicAdd(base + 0, v.x);
    atomicAdd(base + 1, v.y);
    atomicAdd(base + 2, v.z);
    atomicAdd(base + 3, v.w);
}

// ---------------------------------------------------------------------------
// TDM helper: async-load a 16-row x 128-dword f32 tile (contiguous rows) into
// LDS with a 4-dword pad after every 128 dwords (LDS row stride = 132 dwords).
// D# encoding per CDNA5 ISA ch.8 (group0/group1; groups 2/3 unused for 2D).
// ---------------------------------------------------------------------------
__device__ __forceinline__ void tdm_load_tile128(const float* gbase,
                                                 unsigned int lds_byte_off) {
    unsigned long long ga = (unsigned long long)(uintptr_t)gbase;
    u32x4 g0;
    g0.x = 1u;                                        // count=1 (valid user D#)
    g0.y = lds_byte_off;                              // lds_addr
    g0.z = (unsigned int)ga;                          // global_addr[31:0]
    g0.w = (unsigned int)((ga >> 32) & 0x01ffffffu)   // global_addr[56:32]
         | (2u << 30);                                // type=2 ("image")
    i32x8 g1;
    g1[0] = (int)((2u << 16)        // data_size = 4B
                | (1u << 20)        // pad_enable
                | (6u << 22)        // pad_interval: 2^(6+1) = 128 dwords
                | (3u << 25));      // pad_amount: 3+1 = 4 dwords
    g1[1] = (int)(128u << 16);      // tensor_dim0[15:0]=128 (atomic_barrier=0)
    g1[2] = (int)(16u << 16);       // tensor_dim0 hi=0 | tensor_dim1 lo=16
    g1[3] = (int)(128u << 16);      // tensor_dim1 hi=0 | tile_dim0=128
    g1[4] = 16;                     // tile_dim1=16, tile_dim2=0
    g1[5] = 128;                    // tensor_dim0_stride = 128 dwords
    g1[6] = 0;                      // stride hi / tensor_dim1_stride lo
    g1[7] = 0;
    i32x4 z4 = {0, 0, 0, 0};
#if defined(__clang_major__) && (__clang_major__ >= 23)
    i32x8 z8 = {0, 0, 0, 0, 0, 0, 0, 0};
    __builtin_amdgcn_tensor_load_to_lds(g0, g1, z4, z4, z8, 0);
#else
    __builtin_amdgcn_tensor_load_to_lds(g0, g1, z4, z4, 0);
#endif
}

// ---------------------------------------------------------------------------
// Shared epilogue: bias + instance-norm + ReLU on a 16x128 tile held in Ot.
// ---------------------------------------------------------------------------
__device__ __forceinline__ void norm_relu_store(
    float* Ot, float* partsum, float* stats,
    int tid, int tile, float* __restrict__ out) {
    const int OS = 132;
    __syncthreads();
    {
        int r = tid >> 3, pp = tid & 7;
        float s = 0.0f, s2 = 0.0f;
#pragma unroll
        for (int j = 0; j < 16; ++j) {
            float v = Ot[r * OS + pp * 16 + j];
            s += v; s2 += v * v;
        }
        partsum[(r * 8 + pp) * 2]     = s;
        partsum[(r * 8 + pp) * 2 + 1] = s2;
    }
    __syncthreads();
    if (tid < 16) {
        float S = 0.0f, S2 = 0.0f;
        for (int j = 0; j < 8; ++j) {
            S  += partsum[(tid * 8 + j) * 2];
            S2 += partsum[(tid * 8 + j) * 2 + 1];
        }
        float m   = S * (1.0f / 128.0f);
        float var = S2 * (1.0f / 128.0f) - m * m;
        stats[tid * 2]     = m;
        stats[tid * 2 + 1] = rsqrtf(var + 1e-5f);
    }
    __syncthreads();
    {
        int r = tid >> 3, pp = tid & 7;
        float m  = stats[r * 2];
        float rs = stats[r * 2 + 1];
        int node = tile * 16 + r;
#pragma unroll
        for (int q = 0; q < 4; ++q) {
            float4 o;
            o.x = fmaxf((Ot[r * OS + pp * 16 + q * 4 + 0] - m) * rs, 0.0f);
            o.y = fmaxf((Ot[r * OS + pp * 16 + q * 4 + 1] - m) * rs, 0.0f);
            o.z = fmaxf((Ot[r * OS + pp * 16 + q * 4 + 2] - m) * rs, 0.0f);
            o.w = fmaxf((Ot[r * OS + pp * 16 + q * 4 + 3] - m) * rs, 0.0f);
            *reinterpret_cast<float4*>(&out[node * HID + pp * 16 + q * 4]) = o;
        }
    }
}

// ---------------------------------------------------------------------------
// Layer-1 conv (CIN=31): manual LDS staging (rows too short for TDM padding),
// f32 WMMA, fused bias + instance-norm + ReLU. One block per 16-node tile.
// ---------------------------------------------------------------------------
__global__ __launch_bounds__(128) void conv_small(
    const float* __restrict__ agg, const float* __restrict__ xin,
    const float* __restrict__ Wf,   // fragment-major, panels=16
    const float* __restrict__ bias,
    float* __restrict__ out) {
    constexpr int CIN = IN_CH, K2 = 2 * CIN, Kp = 64, AS = Kp + 1;
    __shared__ float At[16 * AS];
    __shared__ float Ot[16 * 132];
    __shared__ float partsum[16 * 8 * 2];
    __shared__ float stats[16 * 2];

    const int tid  = threadIdx.x;
    const int lane = tid & 31;
    const int wave = tid >> 5;
    const int lo   = lane & 15;
    const int hi   = lane >> 4;
    const int tile = blockIdx.x;

    for (int i = tid; i < 16 * Kp; i += 128) {
        int r = i / Kp, k = i % Kp;
        int node = tile * 16 + r;
        float v = 0.0f;
        if (k < CIN)      v = agg[node * CIN + k];
        else if (k < K2)  v = xin[node * CIN + (k - CIN)];
        At[r * AS + k] = v;
    }
    __syncthreads();

    const float2* Wf2 = reinterpret_cast<const float2*>(Wf);
    v8f acc0 = {}, acc1 = {};
    const int nt0 = wave, nt1 = wave + 4;
    const int col0 = nt0 * 16 + lo;
    const int col1 = nt1 * 16 + lo;
#pragma unroll 4
    for (int p = 0; p < 16; ++p) {
        const int k0 = (p << 2) + (hi << 1);
        v2f a, b0, b1;
        a[0] = At[lo * AS + k0];
        a[1] = At[lo * AS + k0 + 1];
        float2 w0 = Wf2[(nt0 * 16 + p) * 32 + lane];
        float2 w1 = Wf2[(nt1 * 16 + p) * 32 + lane];
        b0[0] = w0.x; b0[1] = w0.y;
        b1[0] = w1.x; b1[1] = w1.y;
        acc0 = __builtin_amdgcn_wmma_f32_16x16x4_f32(
            false, a, false, b0, (short)0, acc0, false, false);
        acc1 = __builtin_amdgcn_wmma_f32_16x16x4_f32(
            false, a, false, b1, (short)0, acc1, false, false);
    }
    const float bb0 = bias[col0], bb1 = bias[col1];
#pragma unroll
    for (int v = 0; v < 8; ++v) {
        int row = v + hi * 8;
        Ot[row * 132 + col0] = acc0[v] + bb0;
        Ot[row * 132 + col1] = acc1[v] + bb1;
    }
    norm_relu_store(Ot, partsum, stats, tid, tile, out);
}

// ---------------------------------------------------------------------------
// Layers 2/3 conv (CIN=128): grid-stride, double-buffered TDM staging of the
// [agg | x] A-tiles, f32 WMMA, fused bias + instance-norm + ReLU.
// Wave 0 drives the Tensor Data Mover one tile ahead (TENSORcnt pipelining).
// ---------------------------------------------------------------------------
__global__ __launch_bounds__(128) void conv_big(
    const float* __restrict__ agg, const float* __restrict__ xin,
    const float* __restrict__ Wf,   // fragment-major, panels=64
    const float* __restrict__ bias,
    float* __restrict__ out) {
    __shared__ float Abuf[2][2][16 * 132];   // [buffer][agg|x][16 rows x 132]
    __shared__ float Ot[16 * 132];
    __shared__ float partsum[16 * 8 * 2];
    __shared__ float stats[16 * 2];

    const int tid  = threadIdx.x;
    const int lane = tid & 31;
    const int wave = tid >> 5;
    const int lo   = lane & 15;
    const int hi   = lane >> 4;
    const int G    = gridDim.x;
    const int t0   = blockIdx.x;

    const float2* Wf2 = reinterpret_cast<const float2*>(Wf);
    const int nt0 = wave, nt1 = wave + 4;
    const int col0 = nt0 * 16 + lo;
    const int col1 = nt1 * 16 + lo;
    const float bb0 = bias[col0], bb1 = bias[col1];

    unsigned int lA[2], lX[2];
    lA[0] = (unsigned int)(uintptr_t)(void*)&Abuf[0][0][0];
    lX[0] = (unsigned int)(uintptr_t)(void*)&Abuf[0][1][0];
    lA[1] = (unsigned int)(uintptr_t)(void*)&Abuf[1][0][0];
    lX[1] = (unsigned int)(uintptr_t)(void*)&Abuf[1][1][0];

    int cur = 0;
    if (wave == 0) {  // prime the pipeline: stage first tile into buffer 0
        tdm_load_tile128(agg + (long long)t0 * 16 * HID, lA[0]);
        tdm_load_tile128(xin + (long long)t0 * 16 * HID, lX[0]);
    }
    for (int t = t0; t < N_TILES; t += G) {
        const bool hasnext = (t + G) < N_TILES;
        if (wave == 0) {
            if (hasnext) {
                tdm_load_tile128(agg + (long long)(t + G) * 16 * HID, lA[cur ^ 1]);
                tdm_load_tile128(xin + (long long)(t + G) * 16 * HID, lX[cur ^ 1]);
                __builtin_amdgcn_s_wait_tensorcnt(2);  // current tile's 2 TDM ops done
            } else {
                __builtin_amdgcn_s_wait_tensorcnt(0);
            }
        }
        __syncthreads();   // A-tiles visible to all waves

        const float* Aagg = &Abuf[cur][0][0];
        const float* Ax   = &Abuf[cur][1][0];
        v8f acc0 = {}, acc1 = {};
#pragma unroll 4
        for (int p = 0; p < 32; ++p) {   // K panels from agg half
            const int k0 = (p << 2) + (hi << 1);
            const float2 av = *reinterpret_cast<const float2*>(&Aagg[lo * 132 + k0]);
            float2 w0 = Wf2[(nt0 * 64 + p) * 32 + lane];
            float2 w1 = Wf2[(nt1 * 64 + p) * 32 + lane];
            v2f a, b0, b1;
            a[0] = av.x; a[1] = av.y;
            b0[0] = w0.x; b0[1] = w0.y;
            b1[0] = w1.x; b1[1] = w1.y;
            acc0 = __builtin_amdgcn_wmma_f32_16x16x4_f32(
                false, a, false, b0, (short)0, acc0, false, false);
            acc1 = __builtin_amdgcn_wmma_f32_16x16x4_f32(
                false, a, false, b1, (short)0, acc1, false, false);
        }
#pragma unroll 4
        for (int p = 32; p < 64; ++p) {  // K panels from x half
            const int k0 = ((p - 32) << 2) + (hi << 1);
            const float2 av = *reinterpret_cast<const float2*>(&Ax[lo * 132 + k0]);
            float2 w0 = Wf2[(nt0 * 64 + p) * 32 + lane];
            float2 w1 = Wf2[(nt1 * 64 + p) * 32 + lane];
            v2f a, b0, b1;
            a[0] = av.x; a[1] = av.y;
            b0[0] = w0.x; b0[1] = w0.y;
            b1[0] = w1.x; b1[1] = w1.y;
            acc0 = __builtin_amdgcn_wmma_f32_16x16x4_f32(
                false, a, false, b0, (short)0, acc0, false, false);
            acc1 = __builtin_amdgcn_wmma_f32_16x16x4_f32(
                false, a, false, b1, (short)0, acc1, false, false);
        }
#pragma unroll
        for (int v = 0; v < 8; ++v) {
            int row = v + hi * 8;
            Ot[row * 132 + col0] = acc0[v] + bb0;
            Ot[row * 132 + col1] = acc1[v] + bb1;
        }
        norm_relu_store(Ot, partsum, stats, tid, t, out);
        __syncthreads();   // tile fully consumed before buffers are reused
        cur ^= 1;
    }
}

// ---------------------------------------------------------------------------
// Segment-max pooling; batch is monotone => graph g owns a contiguous range.
// ---------------------------------------------------------------------------
__global__ __launch_bounds__(128) void pool_max(
    const float* __restrict__ x1, const float* __restrict__ x2,
    const float* __restrict__ x3, float* __restrict__ h) {
    int g = blockIdx.x;
    int c = threadIdx.x;
    int start = (g * N_NODES + N_GRAPHS - 1) / N_GRAPHS;
    int end   = ((g + 1) * N_NODES + N_GRAPHS - 1) / N_GRAPHS;
    float m1 = -INFINITY, m2 = -INFINITY, m3 = -INFINITY;
    for (int i = start; i < end; ++i) {
        m1 = fmaxf(m1, x1[i * HID + c]);
        m2 = fmaxf(m2, x2[i * HID + c]);
        m3 = fmaxf(m3, x3[i * HID + c]);
    }
    h[g * 384 + c]       = m1;
    h[g * 384 + 128 + c] = m2;
    h[g * 384 + 256 + c] = m3;
}

// ---------------------------------------------------------------------------
// Per-graph MLP head: relu(h@W1T+b1) @ W2T + b2, then row L2-normalize.
// ---------------------------------------------------------------------------
__global__ __launch_bounds__(256) void mlp_head(
    const float* __restrict__ h,
    const float* __restrict__ l1T,   // [384][256]
    const float* __restrict__ l1b,
    const float* __restrict__ l2T,   // [256][64]
    const float* __restrict__ l2b,
    float* __restrict__ out) {
    __shared__ float hr[384];
    __shared__ float t1[256];
    __shared__ float o[64];
    __shared__ float nrm;
    int g = blockIdx.x, tid = threadIdx.x;
    for (int i = tid; i < 384; i += 256) hr[i] = h[g * 384 + i];
    __syncthreads();
    float s = l1b[tid];
    for (int k = 0; k < 384; ++k) s += hr[k] * l1T[k * 256 + tid];
    t1[tid] = fmaxf(s, 0.0f);
    __syncthreads();
    if (tid < 64) {
        float s2 = l2b[tid];
        for (int k = 0; k < 256; ++k) s2 += t1[k] * l2T[k * 64 + tid];
        o[tid] = s2;
    }
    __syncthreads();
    if (tid == 0) {
        float acc = 0.0f;
        for (int j = 0; j < 64; ++j) acc += o[j] * o[j];
        nrm = fmaxf(sqrtf(acc), 1e-12f);
    }
    __syncthreads();
    if (tid < 64) out[g * 64 + tid] = o[tid] / nrm;
}

// ---------------------------------------------------------------------------
// Host side
// ---------------------------------------------------------------------------
extern "C" void kernel_launch(void* const* d_in, const int* in_sizes, int n_in,
                              void* d_out, int out_size, void* d_ws, size_t ws_size,
                              hipStream_t stream) {
    const float* x       = (const float*)d_in[0];   // [50000,31]
    const int*   eidx    = (const int*)d_in[1];     // [2,600000]
    const float* W1_rel  = (const float*)d_in[3];
    const float* b1      = (const float*)d_in[4];
    const float* W1_root = (const float*)d_in[5];
    const float* W2_rel  = (const float*)d_in[6];
    const float* b2      = (const float*)d_in[7];
    const float* W2_root = (const float*)d_in[8];
    const float* W3_rel  = (const float*)d_in[9];
    const float* b3      = (const float*)d_in[10];
    const float* W3_root = (const float*)d_in[11];
    const float* lin1_w  = (const float*)d_in[12];  // [256,384]
    const float* lin1_b  = (const float*)d_in[13];
    const float* lin2_w  = (const float*)d_in[14];  // [64,256]
    const float* lin2_b  = (const float*)d_in[15];
    float* out = (float*)d_out;

    const int* e_src = eidx;
    const int* e_dst = eidx + N_EDGES;

    // workspace layout (floats)
    float* ws = (float*)d_ws;
    float* x1  = ws;                          // 50000*128
    float* x2  = x1  + N_NODES * HID;
    float* x3  = x2  + N_NODES * HID;
    float* agg = x3  + N_NODES * HID;         // max 50000*128
    float* Wf1 = agg + N_NODES * HID;         // 8*16*64  = 8192
    float* Wf2 = Wf1 + 8 * 16 * 64;           // 8*64*64  = 32768
    float* Wf3 = Wf2 + 8 * 64 * 64;
    float* l1T = Wf3 + 8 * 64 * 64;           // 384*256
    float* l2T = l1T + 384 * 256;             // 256*64
    float* h   = l2T + 256 * 64;              // 512*384

    auto cdiv = [](long long a, long long b) { return (int)((a + b - 1) / b); };

    // --- prep: fragment-major conv weights + transposed MLP weights ---
    build_wfrag<<<cdiv(8 * 16 * 64, 256), 256, 0, stream>>>(W1_rel, W1_root, Wf1, IN_CH, 16);
    build_wfrag<<<cdiv(8 * 64 * 64, 256), 256, 0, stream>>>(W2_rel, W2_root, Wf2, HID, 64);
    build_wfrag<<<cdiv(8 * 64 * 64, 256), 256, 0, stream>>>(W3_rel, W3_root, Wf3, HID, 64);
    transpose_w<<<cdiv(384 * 256, 256), 256, 0, stream>>>(lin1_w, l1T, 256, 384);
    transpose_w<<<cdiv(256 * 64, 256), 256, 0, stream>>>(lin2_w, l2T, 64, 256);

    // --- layer 1 (CIN = 31) ---
    zero_f32<<<cdiv(N_NODES * IN_CH, 256), 256, 0, stream>>>(agg, N_NODES * IN_CH);
    scatter_add31<<<cdiv((long long)N_EDGES * IN_CH, 256), 256, 0, stream>>>(
        x, e_src, e_dst, agg);
    conv_small<<<N_TILES, 128, 0, stream>>>(agg, x, Wf1, b1, x1);

    // --- layer 2 (CIN = 128) ---
    zero_f32<<<cdiv(N_NODES * HID, 256), 256, 0, stream>>>(agg, N_NODES * HID);
    scatter_add128<<<cdiv((long long)N_EDGES * 32, 256), 256, 0, stream>>>(
        x1, e_src, e_dst, agg);
    conv_big<<<BIG_GRID, 128, 0, stream>>>(agg, x1, Wf2, b2, x2);

    // --- layer 3 (CIN = 128) ---
    zero_f32<<<cdiv(N_NODES * HID, 256), 256, 0, stream>>>(agg, N_NODES * HID);
    scatter_add128<<<cdiv((long long)N_EDGES * 32, 256), 256, 0, stream>>>(
        x2, e_src, e_dst, agg);
    conv_big<<<BIG_GRID, 128, 0, stream>>>(agg, x2, Wf3, b3, x3);

    // --- pooling + head ---
    pool_max<<<N_GRAPHS, 128, 0, stream>>>(x1, x2, x3, h);
    mlp_head<<<N_GRAPHS, 256, 0, stream>>>(h, l1T, lin1_b, l2T, lin2_b, out);
}